// GSoPAttention_53120155517299
// MI455X (gfx1250) — compile-verified
//
#include <hip/hip_runtime.h>
#include <hip/hip_bf16.h>

#define CH_ 2048
#define DD  16
#define MM  784    // 28*28
#define BB  32
#define KC  256    // K chunk staged in LDS for the conv GEMM

typedef __attribute__((ext_vector_type(2))) float v2f;
typedef __attribute__((ext_vector_type(8))) float v8f;

__device__ __forceinline__ v8f wmma_f32_4(v2f a, v2f b, v8f c) {
  // D = A(16x4 f32) * B(4x16 f32) + C(16x16 f32)
  return __builtin_amdgcn_wmma_f32_16x16x4_f32(
      /*neg_a=*/false, a, /*neg_b=*/false, b,
      /*c_mod=*/(short)0, c, /*reuse_a=*/false, /*reuse_b=*/false);
}

// ---------------------------------------------------------------------------
// k1: Y[b,d,m] = sum_c w_conv[d,c] * x[b,c,m]
// grid: 196 blocks x 256 threads (8 waves/block); 1 wave = one 16-column tile
// ---------------------------------------------------------------------------
__global__ void k1_conv(const float* __restrict__ x,
                        const float* __restrict__ w,
                        float* __restrict__ Y) {
  __shared__ float lds_a[DD * KC];   // 16 KB: w_conv chunk [16][KC]
  const int tid  = threadIdx.x;
  const int lane = tid & 31;
  const int wave = tid >> 5;
  const int t    = blockIdx.x * 8 + wave;      // tile id in [0,1568)
  const int b    = t / 49;
  const int m0   = (t % 49) * 16;
  const int half = lane >> 4;                  // 0: K+{0,1}, 1: K+{2,3}
  const int l    = lane & 15;
  const float* Xb = x + (size_t)b * CH_ * MM;

  v8f acc = {0.f,0.f,0.f,0.f,0.f,0.f,0.f,0.f};

  for (int kc = 0; kc < CH_; kc += KC) {
    // cooperative, k-coalesced load of the A chunk
    for (int i = tid; i < DD * KC; i += 256) {
      int ch = i / KC;
      int kk = i - ch * KC;
      lds_a[i] = w[ch * CH_ + kc + kk];
    }
    __syncthreads();
    #pragma unroll 4
    for (int k = 0; k < KC; k += 4) {
      const int ka = k + 2 * half;             // even -> 8B aligned in LDS
      v2f a = *(const v2f*)&lds_a[l * KC + ka];
      v2f bf;
      const float* xp = Xb + (size_t)(kc + ka) * MM + m0 + l;
      bf.x = xp[0];       // B[ka  ][m0+l]
      bf.y = xp[MM];      // B[ka+1][m0+l]
      acc = wmma_f32_4(a, bf, acc);
    }
    __syncthreads();
  }

  // D layout: VGPR r -> M = r + 8*half, lane l -> N
  float* Yb = Y + (size_t)b * DD * MM;
  #pragma unroll
  for (int r = 0; r < 8; ++r)
    Yb[(r + 8 * half) * MM + m0 + l] = acc[r];
}

// ---------------------------------------------------------------------------
// k2: per-channel mean / rstd over (B, M).  grid: 16 blocks x 256
// stats[0..15]=mu, stats[16..31]=rsqrt(var+eps)
// ---------------------------------------------------------------------------
__global__ void k2_stats(const float* __restrict__ Y, float* __restrict__ stats) {
  __shared__ float s1[256], s2[256];
  const int ch = blockIdx.x, tid = threadIdx.x;
  float a1 = 0.f, a2 = 0.f;
  for (int i = tid; i < BB * MM; i += 256) {
    int b = i / MM, m = i - b * MM;
    float v = Y[((size_t)b * DD + ch) * MM + m];
    a1 += v; a2 += v * v;
  }
  s1[tid] = a1; s2[tid] = a2;
  __syncthreads();
  for (int s = 128; s > 0; s >>= 1) {
    if (tid < s) { s1[tid] += s1[tid + s]; s2[tid] += s2[tid + s]; }
    __syncthreads();
  }
  if (tid == 0) {
    const float n  = (float)(BB * MM);
    const float mu = s1[0] / n;
    stats[ch]      = mu;
    stats[DD + ch] = rsqrtf(s2[0] / n - mu * mu + 1e-5f);
  }
}

// ---------------------------------------------------------------------------
// k3: BN+ReLU, spatial centering, cov = Yc*Yc^T / M via WMMA.
// grid: 32 blocks (one per batch) x 256 threads; wave 0 runs the WMMA loop.
// ---------------------------------------------------------------------------
__global__ void k3_cov(const float* __restrict__ Y,
                       const float* __restrict__ stats,
                       const float* __restrict__ g1,
                       const float* __restrict__ b1,
                       float* __restrict__ cov) {
  __shared__ float ybuf[DD * MM];   // ~50 KB
  __shared__ float red[256];
  __shared__ float cmean[DD];
  const int b = blockIdx.x, tid = threadIdx.x;
  const float* Yb = Y + (size_t)b * DD * MM;

  for (int i = tid; i < DD * MM; i += 256) {
    int ch = i / MM;
    float v = g1[ch] * (Yb[i] - stats[ch]) * stats[DD + ch] + b1[ch];
    ybuf[i] = v > 0.f ? v : 0.f;
  }
  __syncthreads();

  { // per-channel spatial mean
    int ch = tid >> 4, sub = tid & 15;
    float p = 0.f;
    for (int m = sub; m < MM; m += 16) p += ybuf[ch * MM + m];
    red[tid] = p;
  }
  __syncthreads();
  if (tid < DD) {
    float s = 0.f;
    #pragma unroll
    for (int j = 0; j < 16; ++j) s += red[tid * 16 + j];
    cmean[tid] = s * (1.0f / (float)MM);
  }
  __syncthreads();
  for (int i = tid; i < DD * MM; i += 256) ybuf[i] -= cmean[i / MM];
  __syncthreads();

  if (tid < 32) {                       // wave 0, EXEC all-ones
    const int half = tid >> 4, l = tid & 15;
    v8f acc = {0.f,0.f,0.f,0.f,0.f,0.f,0.f,0.f};
    for (int k = 0; k < MM; k += 4) {
      const int ka = k + 2 * half;      // even -> ds_load_b64
      v2f f = *(const v2f*)&ybuf[l * MM + ka];
      acc = wmma_f32_4(f, f, acc);      // symmetric: A-frag == B-frag
    }
    float* cb = cov + (size_t)b * DD * DD;
    #pragma unroll
    for (int r = 0; r < 8; ++r)
      cb[(r + 8 * half) * DD + l] = acc[r] * (1.0f / (float)MM);
  }
}

// ---------------------------------------------------------------------------
// k4: row-BN over (b, axis2) per group + grouped row conv -> r[32,64]
// single block of 256 threads
// ---------------------------------------------------------------------------
__global__ void k4_row(const float* __restrict__ cov,
                       const float* __restrict__ g2,
                       const float* __restrict__ b2,
                       const float* __restrict__ w_row,
                       float* __restrict__ r_out) {
  __shared__ float c[BB * DD * DD];     // 32 KB
  __shared__ float p1[256], p2[256];
  __shared__ float mu2[DD], rs2[DD];
  const int tid = threadIdx.x;
  for (int i = tid; i < BB * DD * DD; i += 256) c[i] = cov[i];
  __syncthreads();

  { // stats per group g over (b, h)
    int g = tid >> 4, sub = tid & 15;
    float a1 = 0.f, a2 = 0.f;
    for (int bb = sub; bb < BB; bb += 16)
      for (int h = 0; h < DD; ++h) {
        float v = c[bb * DD * DD + g * DD + h];
        a1 += v; a2 += v * v;
      }
    p1[tid] = a1; p2[tid] = a2;
  }
  __syncthreads();
  if (tid < DD) {
    float s1 = 0.f, s2 = 0.f;
    #pragma unroll
    for (int j = 0; j < 16; ++j) { s1 += p1[tid * 16 + j]; s2 += p2[tid * 16 + j]; }
    const float n  = (float)(BB * DD);
    const float mu = s1 / n;
    mu2[tid] = mu;
    rs2[tid] = rsqrtf(s2 / n - mu * mu + 1e-5f);
  }
  __syncthreads();

  for (int i = tid; i < BB * 4 * DD; i += 256) {   // 2048 outputs
    int bb = i >> 6;
    int gj = i & 63;
    int g  = gj >> 2;
    float s = 0.f;
    #pragma unroll
    for (int h = 0; h < DD; ++h) {
      float z = g2[g] * (c[bb * DD * DD + g * DD + h] - mu2[g]) * rs2[g] + b2[g];
      s += w_row[gj * DD + h] * z;
    }
    r_out[i] = s;
  }
}

// ---------------------------------------------------------------------------
// k5: out[b,o] = sigmoid( w_fc[o,:] . r[b,:] ).  grid: 256 blocks x 256
// ---------------------------------------------------------------------------
__global__ void k5_fc(const float* __restrict__ r,
                      const float* __restrict__ w_fc,
                      float* __restrict__ out) {
  __shared__ float rb[64];
  const int b  = blockIdx.x >> 3;
  const int o0 = (blockIdx.x & 7) * 256;
  const int tid = threadIdx.x;
  if (tid < 64) rb[tid] = r[b * 64 + tid];
  __syncthreads();
  const int o = o0 + tid;
  const float* wr = w_fc + (size_t)o * 64;
  float s = 0.f;
  #pragma unroll 8
  for (int cc = 0; cc < 64; ++cc) s += wr[cc] * rb[cc];
  out[(size_t)b * CH_ + o] = 1.0f / (1.0f + __expf(-s));
}

// ---------------------------------------------------------------------------
extern "C" void kernel_launch(void* const* d_in, const int* in_sizes, int n_in,
                              void* d_out, int out_size, void* d_ws, size_t ws_size,
                              hipStream_t stream) {
  const float* x      = (const float*)d_in[0];
  const float* w_conv = (const float*)d_in[1];
  const float* g1     = (const float*)d_in[2];
  const float* b1     = (const float*)d_in[3];
  const float* g2     = (const float*)d_in[4];
  const float* b2     = (const float*)d_in[5];
  const float* w_row  = (const float*)d_in[6];
  const float* w_fc   = (const float*)d_in[7];
  float* out = (float*)d_out;

  // workspace layout (floats)
  float* Y     = (float*)d_ws;                    // 32*16*784 = 401408
  float* stats = Y + (size_t)BB * DD * MM;        // 32
  float* cov   = stats + 32;                      // 32*16*16 = 8192
  float* r     = cov + (size_t)BB * DD * DD;      // 32*64    = 2048

  k1_conv <<<196, 256, 0, stream>>>(x, w_conv, Y);
  k2_stats<<<16,  256, 0, stream>>>(Y, stats);
  k3_cov  <<<32,  256, 0, stream>>>(Y, stats, g1, b1, cov);
  k4_row  <<<1,   256, 0, stream>>>(cov, g2, b2, w_row, r);
  k5_fc   <<<256, 256, 0, stream>>>(r, w_fc, out);
}